// OTContrastiveLoss_53566832116252
// MI455X (gfx1250) — compile-verified
//
#include <hip/hip_runtime.h>
#include <hip/hip_bf16.h>

// ---------------------------------------------------------------------------
// OT contrastive loss on gfx1250 (MI455X).
// Shapes: N=17, C=256, H=W=24 -> HW=576, NT=16*576=9216.
// ---------------------------------------------------------------------------

#define HW      576
#define NT      9216
#define CH      256
#define IMGSZ   147456      // C*H*W = 256*576
#define MELEMS  5308416     // HW*NT
#define TELEMS  2359296     // NT*CH
#define RELEMS  147456      // HW*CH
#define ITERS   500
#define MARGIN  0.7f

typedef float v2f __attribute__((ext_vector_type(2)));
typedef float v8f __attribute__((ext_vector_type(8)));

// --------------------------- T gather --------------------------------------
// target_features = x[1:].reshape(16,256,576).transpose(1,0,2).reshape(-1,256)
// T flat index F maps to source x[(1+n)][c0][hw], F = c0*9216 + n*576 + hw.
__global__ void ot_gatherT(const float* __restrict__ x, float* __restrict__ T) {
    unsigned F = blockIdx.x * 256u + threadIdx.x;           // 9216 blocks exact
    unsigned c0 = F / 9216u;
    unsigned r  = F - c0 * 9216u;
    unsigned n  = r / 576u;
    unsigned hw = r - n * 576u;
    T[F] = x[(unsigned)IMGSZ * (1u + n) + c0 * 576u + hw];
}

// --------------------------- log(att), v=0 ---------------------------------
__global__ void ot_prep(const float* __restrict__ att,
                        float* __restrict__ lmu, float* __restrict__ lnu,
                        float* __restrict__ v) {
    int idx = blockIdx.x * 256 + threadIdx.x;
    if (idx < HW) lmu[idx] = __logf(att[idx]);
    int j = idx - HW;
    if (j >= 0 && j < NT) { lnu[j] = __logf(att[HW + j]); v[j] = 0.0f; }
}

// --------------------------- row norms -------------------------------------
// blocks [0,576): |Q_i|^2 ; blocks [576, 576+9216): |T_j|^2
__global__ void ot_norms(const float* __restrict__ Q, const float* __restrict__ T,
                         float* __restrict__ qn, float* __restrict__ tn) {
    __shared__ float sred[256];
    int b = blockIdx.x, t = threadIdx.x;
    const float* row = (b < HW) ? (Q + b * CH) : (T + (b - HW) * CH);
    float vv = row[t];
    sred[t] = vv * vv;
    __syncthreads();
    for (int s = 128; s > 0; s >>= 1) {
        if (t < s) sred[t] += sred[t + s];
        __syncthreads();
    }
    if (t == 0) { if (b < HW) qn[b] = sred[0]; else tn[b - HW] = sred[0]; }
}

// --------------------------- GEMM1: cost matrix ----------------------------
// M[i,j] = qn[i] + tn[j] - 2 * sum_k Q[i,k]*T[j,k]   via v_wmma_f32_16x16x4_f32
// One wave per 16x16 tile; K=256 in steps of 4.
__global__ void ot_gemm_M(const float* __restrict__ Q, const float* __restrict__ T,
                          const float* __restrict__ qn, const float* __restrict__ tn,
                          float* __restrict__ M) {
    int wave = (blockIdx.x * blockDim.x + threadIdx.x) >> 5;   // 20736 waves exact
    int lane = threadIdx.x & 31;
    int tile_m = wave % 36;          // 36*16 = 576 rows
    int tile_n = wave / 36;          // 576*16 = 9216 cols
    int row0 = tile_m << 4, col0 = tile_n << 4;
    int m  = lane & 15;              // A row / B col / D col
    int kh = lane >> 4;              // K half-select
    const float* qrow = Q + (row0 + m) * CH + 2 * kh;          // A: Q rows
    const float* trow = T + (col0 + m) * CH + 2 * kh;          // B = T^T: T rows
    v8f acc = {};
    for (int kb = 0; kb < CH; kb += 4) {
        v2f a = *(const v2f*)(qrow + kb);
        v2f b = *(const v2f*)(trow + kb);
        acc = __builtin_amdgcn_wmma_f32_16x16x4_f32(
            false, a, false, b, (short)0, acc, false, false);
    }
    int col = col0 + m;
    float tnc = tn[col];
#pragma unroll
    for (int r = 0; r < 8; ++r) {
        int row = row0 + r + 8 * kh;
        M[row * NT + col] = qn[row] + tnc - 2.0f * acc[r];
    }
}

// --------------------------- Sinkhorn u-update -----------------------------
// u[i] = lmu[i] - logsumexp_j( M[i,j] + v[j] ).  One block (256 thr) per row.
__global__ void ot_u_update(const float* __restrict__ M, const float* __restrict__ v,
                            const float* __restrict__ lmu, float* __restrict__ u) {
    __shared__ float sm[256], ss[256];
    int i = blockIdx.x, t = threadIdx.x;
    const float* mrow = M + i * NT;
    float mx = -3.0e38f, s = 0.0f;
    for (int j = t; j < NT; j += 256) {                 // 36 elems/thread, coalesced
        float a = mrow[j] + v[j];
        if (a > mx) { s = s * __expf(mx - a) + 1.0f; mx = a; }
        else        { s += __expf(a - mx); }
    }
    sm[t] = mx; ss[t] = s;
    __syncthreads();
    for (int st = 128; st > 0; st >>= 1) {
        if (t < st) {
            float m1 = sm[t], s1 = ss[t], m2 = sm[t + st], s2 = ss[t + st];
            if (m1 >= m2) { ss[t] = s1 + s2 * __expf(m2 - m1); }
            else          { sm[t] = m2; ss[t] = s2 + s1 * __expf(m1 - m2); }
        }
        __syncthreads();
    }
    if (t == 0) u[i] = lmu[i] - (sm[0] + __logf(ss[0]));
}

// --------------------------- Sinkhorn v-update -----------------------------
// v[j] = lnu[j] - logsumexp_i( M[i,j] + u[i] ).  One thread per column.
__global__ void ot_v_update(const float* __restrict__ M, const float* __restrict__ u,
                            const float* __restrict__ lnu, float* __restrict__ v) {
    int j = blockIdx.x * 256 + threadIdx.x;             // 36 blocks exact
    const float* col = M + j;
    float mx = -3.0e38f, s = 0.0f;
    for (int i = 0; i < HW; ++i) {
        float a = col[i * NT] + u[i];                   // coalesced across lanes
        if (a > mx) { s = s * __expf(mx - a) + 1.0f; mx = a; }
        else        { s += __expf(a - mx); }
    }
    v[j] = lnu[j] - (mx + __logf(s));
}

// --------------------------- P = exp(M + u + v) in place -------------------
__global__ void ot_expP(float* __restrict__ M, const float* __restrict__ u,
                        const float* __restrict__ v) {
    int idx = blockIdx.x * 256 + threadIdx.x;           // 20736 blocks exact
    int i = idx / NT, j = idx - i * NT;
    M[idx] = __expf(M[idx] + u[i] + v[j]);
}

// --------------------------- GEMM2: R = P @ T ------------------------------
// (576x9216)@(9216x256), K=9216 in steps of 4.  One wave per 16x16 tile.
__global__ void ot_gemm_R(const float* __restrict__ P, const float* __restrict__ T,
                          float* __restrict__ R) {
    int wave = (blockIdx.x * blockDim.x + threadIdx.x) >> 5;   // 576 waves exact
    int lane = threadIdx.x & 31;
    int tile_m = wave % 36;          // 576 rows
    int tile_n = wave / 36;          // 256 cols -> 16 tiles
    int row0 = tile_m << 4, col0 = tile_n << 4;
    int m  = lane & 15;
    int kh = lane >> 4;
    const float* prow = P + (row0 + m) * NT + 2 * kh;
    const float* tcol = T + col0 + m;                   // B[k][n] = T[k*256 + n]
    v8f acc = {};
    for (int kb = 0; kb < NT; kb += 4) {
        if ((kb & 255) == 0)
            __builtin_prefetch(prow + kb + 256, 0, 1);  // global_prefetch_b8
        v2f a = *(const v2f*)(prow + kb);
        v2f b;
        b.x = tcol[(kb + 2 * kh)     * CH];
        b.y = tcol[(kb + 2 * kh + 1) * CH];
        acc = __builtin_amdgcn_wmma_f32_16x16x4_f32(
            false, a, false, b, (short)0, acc, false, false);
    }
#pragma unroll
    for (int r = 0; r < 8; ++r)
        R[(row0 + r + 8 * kh) * CH + col0 + m] = acc[r];
}

// --------------------------- per-row distance^2 ----------------------------
__global__ void ot_dist(const float* __restrict__ Q, const float* __restrict__ att,
                        const float* __restrict__ R, float* __restrict__ d2) {
    __shared__ float sred[256];
    int i = blockIdx.x, t = threadIdx.x;
    float diff = att[i] * Q[i * CH + t] - R[i * CH + t];
    sred[t] = diff * diff;
    __syncthreads();
    for (int s = 128; s > 0; s >>= 1) {
        if (t < s) sred[t] += sred[t + s];
        __syncthreads();
    }
    if (t == 0) d2[i] = sred[0];
}

// --------------------------- final contrastive reduction -------------------
// loss = 0.5*L*S1 + 0.5*(NT-L)*S2,  L = 576*sum(label[1:]),
// S1 = sum d^2, S2 = sum relu(MARGIN-d)^2.
__global__ void ot_final(const float* __restrict__ d2, const float* __restrict__ label,
                         float* __restrict__ out) {
    __shared__ float r1[256], r2[256];
    int t = threadIdx.x;
    float s1 = 0.0f, s2 = 0.0f;
    for (int i = t; i < HW; i += 256) {
        float dd = d2[i];
        s1 += dd;
        float d = __builtin_sqrtf(dd);
        float h = MARGIN - d;
        if (h > 0.0f) s2 += h * h;
    }
    r1[t] = s1; r2[t] = s2;
    __syncthreads();
    for (int s = 128; s > 0; s >>= 1) {
        if (t < s) { r1[t] += r1[t + s]; r2[t] += r2[t + s]; }
        __syncthreads();
    }
    if (t == 0) {
        float ls = 0.0f;
        for (int k = 1; k < 17; ++k) ls += label[k];
        float L = ls * (float)HW;
        out[0] = 0.5f * L * r1[0] + 0.5f * ((float)NT - L) * r2[0];
    }
}

// ---------------------------------------------------------------------------
extern "C" void kernel_launch(void* const* d_in, const int* in_sizes, int n_in,
                              void* d_out, int out_size, void* d_ws, size_t ws_size,
                              hipStream_t stream) {
    const float* x     = (const float*)d_in[0];   // (17,256,24,24)
    const float* att   = (const float*)d_in[1];   // (17,24,24) flat
    const float* label = (const float*)d_in[2];   // (17,)
    float* out = (float*)d_out;

    // Workspace layout (floats)
    float* ws  = (float*)d_ws;
    float* T   = ws;                        // 2,359,296
    float* M   = T   + TELEMS;              // 5,308,416
    float* R   = M   + MELEMS;              //   147,456
    float* qn  = R   + RELEMS;              //       576
    float* tn  = qn  + HW;                  //     9,216
    float* lmu = tn  + NT;                  //       576
    float* lnu = lmu + HW;                  //     9,216
    float* u   = lnu + NT;                  //       576
    float* v   = u   + HW;                  //     9,216
    float* d2  = v   + NT;                  //       576

    const float* Q = x;                     // x[0].reshape(-1,256) is a flat view

    // 1) gather T, logs, v=0, norms
    ot_gatherT<<<TELEMS / 256, 256, 0, stream>>>(x, T);
    ot_prep<<<(HW + NT + 255) / 256, 256, 0, stream>>>(att, lmu, lnu, v);
    ot_norms<<<HW + NT, 256, 0, stream>>>(Q, T, qn, tn);

    // 2) cost matrix via WMMA (20736 waves, 4 waves/block)
    ot_gemm_M<<<(36 * (NT / 16)) / 4, 128, 0, stream>>>(Q, T, qn, tn, M);

    // 3) Sinkhorn: 500 iterations, stream-ordered device-wide sync
    for (int it = 0; it < ITERS; ++it) {
        ot_u_update<<<HW, 256, 0, stream>>>(M, v, lmu, u);
        ot_v_update<<<NT / 256, 256, 0, stream>>>(M, u, lnu, v);
    }

    // 4) transport plan, apply to targets via WMMA
    ot_expP<<<MELEMS / 256, 256, 0, stream>>>(M, u, v);
    ot_gemm_R<<<(36 * (CH / 16)) / 4, 128, 0, stream>>>(M, T, R);

    // 5) distances + contrastive reduction
    ot_dist<<<HW, 256, 0, stream>>>(Q, att, R, d2);
    ot_final<<<1, 256, 0, stream>>>(d2, label, out);
}